// SimpleGGNN_30356828848478
// MI455X (gfx1250) — compile-verified
//
#include <hip/hip_runtime.h>
#include <hip/hip_bf16.h>

#define N_NODES 100000
#define N_EDGES 3200000
#define IN_C    40
#define OUT_C   32
#define NL      3
#define N_TILES (N_NODES / 16)   // 6250 exactly

typedef __attribute__((ext_vector_type(2))) float v2f;
typedef __attribute__((ext_vector_type(8))) float v8f;
typedef unsigned int v4u __attribute__((ext_vector_type(4)));
typedef int          v4i __attribute__((ext_vector_type(4)));
typedef int          v8i __attribute__((ext_vector_type(8)));

// Tensor Data Mover: arity differs per toolchain (bridge doc):
//   ROCm 7.2 / clang-22 : 5 args (g0, g1, g2, g3, cpol)
//   amdgpu-tc / clang-23: 6 args (g0, g1, g2, g3, g4, cpol)
#if defined(__clang_major__) && (__clang_major__ >= 23)
#define TDM_LOAD_TO_LDS(g0, g1) \
  __builtin_amdgcn_tensor_load_to_lds((g0), (g1), (v4i){}, (v4i){}, (v8i){}, 0)
#else
#define TDM_LOAD_TO_LDS(g0, g1) \
  __builtin_amdgcn_tensor_load_to_lds((g0), (g1), (v4i){}, (v4i){}, 0)
#endif

// ---------------------------------------------------------------------------
// h = x @ lin_w^T + lin_b   (scalar: K=40 not WMMA-friendly; 128M FMA, trivial)
// ---------------------------------------------------------------------------
__global__ void __launch_bounds__(256) ggnn_init_h(
    const float* __restrict__ x, const float* __restrict__ lin_w,
    const float* __restrict__ lin_b, float* __restrict__ h) {
  int t = blockIdx.x * blockDim.x + threadIdx.x;
  if (t >= N_NODES * OUT_C) return;
  int n = t >> 5, o = t & 31;
  const float* xr = x + n * IN_C;
  const float* wr = lin_w + o * IN_C;
  float acc = lin_b[o];
#pragma unroll
  for (int k = 0; k < IN_C; ++k) acc = fmaf(xr[k], wr[k], acc);
  h[t] = acc;
}

// ---------------------------------------------------------------------------
// agg = 0
// ---------------------------------------------------------------------------
__global__ void __launch_bounds__(256) ggnn_zero4(float4* __restrict__ p, int n4) {
  int t = blockIdx.x * blockDim.x + threadIdx.x;
  if (t < n4) p[t] = make_float4(0.f, 0.f, 0.f, 0.f);
}

// ---------------------------------------------------------------------------
// m = h @ W   (W = conv_w[c][i], [32x32] row-major: W[k][o])
// Block = 8 waves = 128 node rows. The block's A slab (128 rows x 128B) is one
// contiguous 16KB region of h -> DMA'd to LDS with TENSOR_LOAD_TO_LDS (D# per
// ISA ch.8; tensor_dim0 clamped to valid remainder so tail OOB reads return 0),
// s_wait_tensorcnt, barrier. Weights pre-packed in LDS as (W[2p][o],W[2p+1][o])
// float2 pairs so every WMMA B fragment is a single aligned ds_load_b64.
// ---------------------------------------------------------------------------
__global__ void __launch_bounds__(256) ggnn_gemm_m(
    const float* __restrict__ h, const float* __restrict__ W,
    float* __restrict__ m) {
  __shared__ __align__(16) float sH[128 * OUT_C];   // 16KB A slab
  __shared__ __align__(16) v2f   sW2[16 * OUT_C];   // 4KB packed B

  // Issue TDM for the A slab (one wave; TDM ignores EXEC, issues once/wave).
  if (threadIdx.x == 0) {
    unsigned long long ga =
        (unsigned long long)(const void*)(h + (size_t)blockIdx.x * 128 * OUT_C);
    int remain = N_NODES * OUT_C - blockIdx.x * (128 * OUT_C);
    unsigned td0 = remain < 128 * OUT_C ? (unsigned)remain : 128u * OUT_C;
    v4u g0;
    g0[0] = 1u;                                          // count=1, user mode
    g0[1] = (unsigned)(size_t)sH;                        // lds_addr (bytes)
    g0[2] = (unsigned)ga;                                // global_addr[31:0]
    g0[3] = ((unsigned)(ga >> 32) & 0x01FFFFFFu) | 0x80000000u;  // [56:32]+type=2
    v8i g1 = {};
    g1[0] = 0x00020000;                    // workgroup_mask=0, data_size=4B
    g1[1] = (int)(td0 << 16);              // tensor_dim0[15:0]
    g1[2] = (int)((td0 >> 16) | (1u << 16));  // tensor_dim0[31:16], tensor_dim1=1
    g1[3] = (int)(4096u << 16);            // tile_dim0 = 4096 elements
    g1[4] = 1;                             // tile_dim1 = 1, tile_dim2 = 0
    g1[5] = (int)td0;                      // tensor_dim0_stride (1 row: unused)
    TDM_LOAD_TO_LDS(g0, g1);
    __builtin_amdgcn_s_wait_tensorcnt((short)0);
  }

  // Pack weights: sW2[p*32+o] = {W[2p][o], W[2p+1][o]}
  for (int i = threadIdx.x; i < 16 * OUT_C; i += 256) {
    int o = i & 31, p = i >> 5;
    v2f w;
    w.x = W[(2 * p) * OUT_C + o];
    w.y = W[(2 * p + 1) * OUT_C + o];
    sW2[i] = w;
  }
  __syncthreads();

  const int lane = threadIdx.x & 31;
  const int wave = threadIdx.x >> 5;
  const int tile = blockIdx.x * 8 + wave;
  if (tile >= N_TILES) return;              // wave-uniform: EXEC stays all-ones
  const int ln = lane & 15;
  const int hi = lane >> 4;
  const v2f* sHv = (const v2f*)sH;
  const int abase = ((wave * 16 + ln) * OUT_C) >> 1;   // lane's row, v2f units

  v8f c0 = {}, c1 = {};
#pragma unroll
  for (int k0 = 0; k0 < OUT_C; k0 += 4) {
    const int p = (k0 >> 1) + hi;                      // pair index = ka/2
    v2f a  = sHv[abase + p];
    v2f b0 = sW2[p * OUT_C + ln];
    v2f b1 = sW2[p * OUT_C + 16 + ln];
    c0 = __builtin_amdgcn_wmma_f32_16x16x4_f32(false, a, false, b0, (short)0, c0, false, false);
    c1 = __builtin_amdgcn_wmma_f32_16x16x4_f32(false, a, false, b1, (short)0, c1, false, false);
  }
  float* mb = m + tile * 16 * OUT_C;
#pragma unroll
  for (int r = 0; r < 8; ++r) {
    mb[(r + 8 * hi) * OUT_C + ln]      = c0[r];
    mb[(r + 8 * hi) * OUT_C + 16 + ln] = c1[r];
  }
}

// ---------------------------------------------------------------------------
// agg[dst] += m[src]    one warp per edge, lane = channel.
// Edge indices are wave-uniform -> scalarized SMEM loads; row gather/atomics
// are 128B-coalesced and L2-resident (38MB << 192MB L2). This is the bound:
// ~1G f32 atomics/launch at L2.
// ---------------------------------------------------------------------------
__global__ void __launch_bounds__(256) ggnn_scatter(
    const int* __restrict__ ei, const float* __restrict__ m,
    float* __restrict__ agg) {
  int t = blockIdx.x * blockDim.x + threadIdx.x;
  int e = t >> 5;
  if (e >= N_EDGES) return;
  int lane = threadIdx.x & 31;
  int s = ei[e];
  int d = ei[N_EDGES + e];
  atomicAdd(&agg[d * OUT_C + lane], m[s * OUT_C + lane]);
}

// ---------------------------------------------------------------------------
// GRU step (fused):  gi = agg @ w_ih^T + b_ih ; gh = h @ w_hh^T + b_hh
//   r = sig(i_r+h_r); z = sig(i_z+h_z); n = tanh(i_n + r*h_n)
//   h = (1-z)*n + z*h        (in place; wave owns its 16 rows exclusively)
// Both 32->96 GEMMs on WMMA f32 (96 wmma/wave). Transposed weights staged in
// LDS pre-packed: sW[p*96+j] = {w[j][2p], w[j][2p+1]} (pairs contiguous in
// global!) -> each B fragment is one aligned ds_load_b64.
// ---------------------------------------------------------------------------
__global__ void __launch_bounds__(256) ggnn_gru(
    const float* __restrict__ agg, float* __restrict__ h,
    const float* __restrict__ w_ih, const float* __restrict__ w_hh,  // [96,32]
    const float* __restrict__ b_ih, const float* __restrict__ b_hh)  // [96]
{
  __shared__ __align__(16) v2f sWi2[16 * 96];   // 12KB
  __shared__ __align__(16) v2f sWh2[16 * 96];   // 12KB
  for (int i = threadIdx.x; i < 16 * 96; i += 256) {
    int p = i & 15, j = i >> 4;
    const float2* wi = (const float2*)(w_ih + j * OUT_C);
    const float2* wh = (const float2*)(w_hh + j * OUT_C);
    float2 a = wi[p];
    float2 b = wh[p];
    v2f va; va.x = a.x; va.y = a.y;
    v2f vb; vb.x = b.x; vb.y = b.y;
    sWi2[p * 96 + j] = va;
    sWh2[p * 96 + j] = vb;
  }
  __syncthreads();

  const int lane = threadIdx.x & 31;
  const int wave = threadIdx.x >> 5;
  const int tile = blockIdx.x * 8 + wave;
  if (tile >= N_TILES) return;              // wave-uniform
  const int ln = lane & 15;
  const int hi = lane >> 4;
  const int row = tile * 16 + ln;

  // Preload this lane's A-fragments for all 8 k-steps (aligned b64 loads).
  const v2f* ar2 = (const v2f*)(agg + row * OUT_C);
  const v2f* hr2 = (const v2f*)(h + row * OUT_C);
  v2f av[8], hv[8];
#pragma unroll
  for (int s = 0; s < 8; ++s) {
    av[s] = ar2[2 * s + hi];
    hv[s] = hr2[2 * s + hi];
  }

  v8f ci[6], ch[6];
#pragma unroll
  for (int ct = 0; ct < 6; ++ct) { ci[ct] = (v8f){}; ch[ct] = (v8f){}; }

#pragma unroll
  for (int ct = 0; ct < 6; ++ct) {
    const int col = ct * 16 + ln;
#pragma unroll
    for (int s = 0; s < 8; ++s) {
      const int p = 2 * s + hi;                       // pair index = ka/2
      v2f bi = sWi2[p * 96 + col];
      ci[ct] = __builtin_amdgcn_wmma_f32_16x16x4_f32(false, av[s], false, bi, (short)0, ci[ct], false, false);
      v2f bh = sWh2[p * 96 + col];
      ch[ct] = __builtin_amdgcn_wmma_f32_16x16x4_f32(false, hv[s], false, bh, (short)0, ch[ct], false, false);
    }
  }

  // Elementwise GRU: output col o = ct*16+ln (ct in {0,1}); o+32 -> ct+2, o+64 -> ct+4
#pragma unroll
  for (int ct = 0; ct < 2; ++ct) {
    const int col = ct * 16 + ln;
    const float bir = b_ih[col], biz = b_ih[col + 32], bin_ = b_ih[col + 64];
    const float bhr = b_hh[col], bhz = b_hh[col + 32], bhn  = b_hh[col + 64];
#pragma unroll
    for (int r = 0; r < 8; ++r) {
      const int ofs = (tile * 16 + r + 8 * hi) * OUT_C + col;
      const float hold = h[ofs];
      const float rg = 1.f / (1.f + expf(-((ci[ct][r] + bir) + (ch[ct][r] + bhr))));
      const float zg = 1.f / (1.f + expf(-((ci[ct + 2][r] + biz) + (ch[ct + 2][r] + bhz))));
      const float ng = tanhf((ci[ct + 4][r] + bin_) + rg * (ch[ct + 4][r] + bhn));
      h[ofs] = (1.f - zg) * ng + zg * hold;
    }
  }
}

// ---------------------------------------------------------------------------
extern "C" void kernel_launch(void* const* d_in, const int* in_sizes, int n_in,
                              void* d_out, int out_size, void* d_ws, size_t ws_size,
                              hipStream_t stream) {
  const float* x      = (const float*)d_in[0];
  const int*   ei     = (const int*)  d_in[1];  // [2, E]
  const float* lin_w  = (const float*)d_in[2];  // [32,40]
  const float* lin_b  = (const float*)d_in[3];  // [32]
  const float* conv_w = (const float*)d_in[4];  // [3,3,32,32]
  const float* w_ih   = (const float*)d_in[5];  // [3,96,32]
  const float* w_hh   = (const float*)d_in[6];  // [3,96,32]
  const float* b_ih   = (const float*)d_in[7];  // [3,96]
  const float* b_hh   = (const float*)d_in[8];  // [3,96]

  float* h   = (float*)d_out;                       // [N,32] lives in d_out
  float* m   = (float*)d_ws;                        // [N,32]
  float* agg = m + (size_t)N_NODES * OUT_C;         // [N,32]

  ggnn_init_h<<<(N_NODES * OUT_C + 255) / 256, 256, 0, stream>>>(x, lin_w, lin_b, h);

  const int gTile = (N_TILES + 7) / 8;              // 782 blocks of 8 waves
  const int n4    = N_NODES * OUT_C / 4;
  const int gScat = (N_EDGES * 32) / 256;           // 400000 blocks

  for (int c = 0; c < NL; ++c) {
    for (int i = 0; i < NL; ++i) {
      ggnn_gemm_m<<<gTile, 256, 0, stream>>>(
          h, conv_w + (size_t)(c * NL + i) * OUT_C * OUT_C, m);
      ggnn_zero4<<<(n4 + 255) / 256, 256, 0, stream>>>((float4*)agg, n4);
      ggnn_scatter<<<gScat, 256, 0, stream>>>(ei, m, agg);
      ggnn_gru<<<gTile, 256, 0, stream>>>(
          agg, h,
          w_ih + (size_t)c * 96 * OUT_C, w_hh + (size_t)c * 96 * OUT_C,
          b_ih + (size_t)c * 96,         b_hh + (size_t)c * 96);
    }
  }
}